// SkipHGNN_82609400971810
// MI455X (gfx1250) — compile-verified
//
#include <hip/hip_runtime.h>
#include <math.h>

#define DS 127
#define DP 128
#define EPSF 1e-7f

typedef __attribute__((ext_vector_type(2))) float v2f;
typedef __attribute__((ext_vector_type(8))) float v8f;

// ---------------- row-wise norm reduction (block = 128 threads) -------------
__device__ __forceinline__ float block_norm128(float v, float* red) {
    const int t = threadIdx.x;
    red[t] = v * v;
    __syncthreads();
    #pragma unroll
    for (int s = 64; s > 0; s >>= 1) {
        if (t < s) red[t] += red[t + s];
        __syncthreads();
    }
    return sqrtf(red[0]);
}

// scale factor of h = logmap0(expmap0(v)) composed in one pass
__device__ __forceinline__ float logexp_scale(float nv) {
    const float n  = fmaxf(nv, EPSF);
    const float ch = coshf(n);
    const float sh = sinhf(n);
    const float ns = fmaxf(sh * nv / n, EPSF);      // ||s|| = sinh(n)*||v||/n
    const float dd = acoshf(fmaxf(ch, 1.0f + EPSF));
    return dd * sh / (n * ns);                      // h_i = scale * v_i
}

__global__ __launch_bounds__(128) void k_prep(const float* __restrict__ x,
                                              float* __restrict__ h) {
    __shared__ float red[128];
    const int row = blockIdx.x, t = threadIdx.x;
    const float v = (t < DS) ? x[(size_t)row * DS + t] : 0.0f;
    const float nv = block_norm128(v, red);
    const float sc = logexp_scale(nv);
    h[(size_t)row * DP + t] = (t < DS) ? v * sc : 0.0f;   // pad col stays 0
}

// h = logmap0(expmap0(relu(agg) + h)) in place (padded layout, col 127 == 0)
__global__ __launch_bounds__(128) void k_update(const float* __restrict__ agg,
                                                float* __restrict__ h) {
    __shared__ float red[128];
    const int row = blockIdx.x, t = threadIdx.x;
    const size_t idx = (size_t)row * DP + t;
    const float v = fmaxf(agg[idx], 0.0f) + h[idx];       // pad lane: 0+0
    const float nv = block_norm128(v, red);
    const float sc = logexp_scale(nv);
    h[idx] = (t < DS) ? v * sc : 0.0f;
}

// out[row] = [cosh(n), sinh(n)*v/n ...]  (exactly 128 outputs per row)
__global__ __launch_bounds__(128) void k_final(const float* __restrict__ h,
                                               float* __restrict__ out) {
    __shared__ float red[128];
    const int row = blockIdx.x, t = threadIdx.x;
    const float v = h[(size_t)row * DP + t];              // col 127 is 0
    const float nv = block_norm128(v, red);
    const float n  = fmaxf(nv, EPSF);
    if (t < DS) out[(size_t)row * DP + 1 + t] = sinhf(n) * v / n;
    else        out[(size_t)row * DP + 0]     = coshf(n); // lane 127 writes t
}

// ------- pack W (127x127 row-major, W[k][n]) into WpT (128x128, N-major) ----
// WpT[n][k] = W[k][n], zero-padded borders: B operand K-pairs become
// contiguous 8-byte loads in the GEMM.
__global__ void k_packWT(const float* __restrict__ W, float* __restrict__ WpT) {
    const int idx = blockIdx.x * blockDim.x + threadIdx.x;
    if (idx >= DP * DP) return;
    const int n = idx >> 7, k = idx & (DP - 1);
    WpT[idx] = (n < DS && k < DS) ? W[k * DS + n] : 0.0f;
}

__global__ void k_zero(float* __restrict__ p, size_t n) {
    const size_t i = (size_t)blockIdx.x * blockDim.x + threadIdx.x;
    if (i < n) p[i] = 0.0f;
}

// ---------------- z = h @ W + b via V_WMMA_F32_16X16X4_F32 ------------------
// One wave per 16x16 output tile; h/z are padded to N16 rows so all loads and
// stores are unconditional (no exec-mask churn around WMMA).
// A layout (16x4 f32): lanes 0-15 -> M=0..15 holding K={kb,kb+1};
//                      lanes16-31 -> same M  holding K={kb+2,kb+3}.
// B layout (4x16 f32): lane%16 = column N; lanes 0-15 hold rows K={kb,kb+1},
//                      lanes16-31 hold rows K={kb+2,kb+3} (one row per VGPR).
// C/D: VGPR r, lanes 0-15 -> M=r, lanes 16-31 -> M=r+8, N = lane%16.
__global__ __launch_bounds__(32) void k_gemm(const float* __restrict__ h,
                                             const float* __restrict__ WpT,
                                             const float* __restrict__ bias,
                                             float* __restrict__ z) {
    const int lane = threadIdx.x;
    const int half = lane >> 4;          // 0 or 1
    const int l16  = lane & 15;
    const int m0   = blockIdx.x * 16;
    const int n0   = blockIdx.y * 16;

    const float* ha = h   + (size_t)(m0 + l16) * DP + 2 * half;  // A stream
    const float* wb = WpT + (size_t)(n0 + l16) * DP + 2 * half;  // B stream

    v8f acc = {};
    #pragma unroll 8
    for (int kb = 0; kb < DP; kb += 4) {
        const v2f a = *(const v2f*)(ha + kb);   // 8B-aligned global_load_b64
        const v2f b = *(const v2f*)(wb + kb);   // 8B-aligned global_load_b64
        acc = __builtin_amdgcn_wmma_f32_16x16x4_f32(
            /*neg_a=*/false, a, /*neg_b=*/false, b,
            /*c_mod=*/(short)0, acc, /*reuse_a=*/false, /*reuse_b=*/false);
    }

    const int   ncol = n0 + l16;
    const float bv   = (ncol < DS) ? bias[ncol] : 0.0f;   // pad col -> +0
    float* zc = z + (size_t)(m0 + half * 8) * DP + ncol;
    #pragma unroll
    for (int r = 0; r < 8; ++r)
        zc[(size_t)r * DP] = acc[r] + bv;                 // rows < N16, in-bounds
}

// ---------------- edge gather + weighted scatter-add (atomics hit L2) -------
__global__ __launch_bounds__(128) void k_edge(const float* __restrict__ z,
                                              const int* __restrict__ src,
                                              const int* __restrict__ dst,
                                              const float* __restrict__ w,
                                              float* __restrict__ agg) {
    const int e = blockIdx.x, t = threadIdx.x;
    const int s = src[e], d = dst[e];
    const float wt = w[e];
    const float v = z[(size_t)s * DP + t] * wt;           // col 127 -> 0
    atomicAdd(&agg[(size_t)d * DP + t], v);               // +0 on pad col
}

// ----------------------------------------------------------------------------
extern "C" void kernel_launch(void* const* d_in, const int* in_sizes, int n_in,
                              void* d_out, int out_size, void* d_ws, size_t ws_size,
                              hipStream_t stream) {
    const float* x  = (const float*)d_in[0];
    const float* W1 = (const float*)d_in[1];
    const float* b1 = (const float*)d_in[2];
    const float* W2 = (const float*)d_in[3];
    const float* b2 = (const float*)d_in[4];
    const int*   es = (const int*)d_in[5];
    const int*   ed = (const int*)d_in[6];
    const float* ew = (const float*)d_in[7];

    const int N   = in_sizes[0] / DS;        // 50000
    const int E   = in_sizes[5];             // 800000
    const int N16 = (N + 15) & ~15;          // rows padded to tile multiple

    float* h   = (float*)d_ws;                   // N16 x 128
    float* z   = h   + (size_t)N16 * DP;         // N16 x 128
    float* agg = z   + (size_t)N16 * DP;         // N16 x 128
    float* WpT = agg + (size_t)N16 * DP;         // 128 x 128 (N-major)

    k_prep<<<N, 128, 0, stream>>>(x, h);

    const float* Ws[2] = {W1, W2};
    const float* bs[2] = {b1, b2};
    for (int layer = 0; layer < 2; ++layer) {
        k_packWT<<<(DP * DP + 255) / 256, 256, 0, stream>>>(Ws[layer], WpT);

        dim3 gg(N16 / 16, DP / 16);
        k_gemm<<<gg, 32, 0, stream>>>(h, WpT, bs[layer], z);

        const size_t an = (size_t)N * DP;
        k_zero<<<(unsigned)((an + 255) / 256), 256, 0, stream>>>(agg, an);

        k_edge<<<E, 128, 0, stream>>>(z, es, ed, ew, agg);

        k_update<<<N, 128, 0, stream>>>(agg, h);
    }

    k_final<<<N, 128, 0, stream>>>(h, (float*)d_out);
}